// SummationMPNN_18365280157746
// MI455X (gfx1250) — compile-verified
//
#include <hip/hip_runtime.h>
#include <hip/hip_bf16.h>

typedef __attribute__((ext_vector_type(2))) float v2f;
typedef __attribute__((ext_vector_type(8))) float v8f;

#define Bm   128
#define Nn   40
#define NF   128
#define EF   16
#define MM   128
#define OUT  128
#define PASSES 4
#define BN   (Bm * Nn)        // 5120 rows, 320 row-tiles of 16

__device__ __forceinline__ v8f wmma4(v2f a, v2f b, v8f c) {
  // D = A(16x4 f32) * B(4x16 f32) + C(16x16 f32)
  return __builtin_amdgcn_wmma_f32_16x16x4_f32(false, a, false, b, (short)0, c, false, false);
}

// ---------------------------------------------------------------------------
// init: hidden = nodes; adjbits[b,i] = bitmask_j(adj[b,i,j]!=0); maskf = any
// ---------------------------------------------------------------------------
__global__ void init_kernel(const int* __restrict__ adj,
                            const float* __restrict__ nodes,
                            float* __restrict__ h0,
                            float* __restrict__ maskf,
                            unsigned long long* __restrict__ adjbits) {
  const int tid = blockIdx.x * blockDim.x + threadIdx.x;
  const int nthr = gridDim.x * blockDim.x;
  for (int idx = tid; idx < BN * NF; idx += nthr)
    h0[idx] = nodes[idx];
  if (tid < BN) {
    const int* arow = adj + tid * Nn;
    unsigned long long bits = 0ull;
    for (int j = 0; j < Nn; ++j)
      if (arow[j] != 0) bits |= (1ull << j);
    adjbits[tid] = bits;
    maskf[tid] = bits ? 1.0f : 0.0f;
  }
}

// ---------------------------------------------------------------------------
// pack_b: repack row-major weight W[K, Ncols] into WMMA-B fragment order:
//   P[(nt*(K/4)+ks)*64 + lane*2 + v] = W[(ks*4 + 2*(lane>>4) + v)*Ncols
//                                        + nt*16 + (lane&15)]
// so each B fragment is one coalesced 8-byte load per lane.
// ---------------------------------------------------------------------------
__global__ void pack_b_kernel(const float* __restrict__ W, float* __restrict__ P,
                              int K, int Ncols) {
  const int total = K * Ncols;
  const int kq = K >> 2;
  for (int idx = blockIdx.x * blockDim.x + threadIdx.x; idx < total;
       idx += gridDim.x * blockDim.x) {
    const int v    = idx & 1;
    const int lane = (idx >> 1) & 31;
    const int t    = idx >> 6;          // nt*(K/4) + ks
    const int ks   = t % kq;
    const int nt   = t / kq;
    const int hi = lane >> 4, lo = lane & 15;
    const int k   = ks * 4 + 2 * hi + v;
    const int col = nt * 16 + lo;
    P[idx] = W[k * Ncols + col];
  }
}

// ---------------------------------------------------------------------------
// proj: A = H @ W1, C = H @ W2 (packed weights). 2560 tiles, 4 waves/block.
// ---------------------------------------------------------------------------
__global__ void __launch_bounds__(128) proj_kernel(const float* __restrict__ X,
                                                   const float* __restrict__ P1,
                                                   const float* __restrict__ P2,
                                                   float* __restrict__ Y1,
                                                   float* __restrict__ Y2) {
  const int lane = threadIdx.x & 31;
  const int wave = threadIdx.x >> 5;
  const int tile = blockIdx.x * 4 + wave;
  const int mt = tile >> 3;
  const int nt = tile & 7;
  const int lo = lane & 15, hi = lane >> 4;
  const int row = mt * 16 + lo;
  const int col = nt * 16 + lo;
  const float* B1 = P1 + (nt * 32) * 64 + lane * 2;   // K/4 = 32 frags
  const float* B2 = P2 + (nt * 32) * 64 + lane * 2;

  v8f acc1 = {};
  v8f acc2 = {};
#pragma unroll 8
  for (int ks = 0; ks < 32; ++ks) {
    v2f a  = *(const v2f*)(X + row * NF + ks * 4 + 2 * hi);
    v2f b1 = *(const v2f*)(B1 + ks * 64);
    v2f b2 = *(const v2f*)(B2 + ks * 64);
    acc1 = wmma4(a, b1, acc1);
    acc2 = wmma4(a, b2, acc2);
  }
#pragma unroll
  for (int r = 0; r < 8; ++r) {
    const int orow = mt * 16 + r + 8 * hi;
    Y1[orow * MM + col] = acc1[r];
    Y2[orow * MM + col] = acc2[r];
  }
}

// ---------------------------------------------------------------------------
// msg: messages[b,i,m] = sum_j adj[b,i,j] * relu(A[b,i,m] + C[b,j,m]
//                                + edges[b,i,j,:]@W3[:,m] + b_msg[m])
// One block per (b,i); 8 waves = 8 m-tiles; j tiled 3x16 (padded past 40).
// ---------------------------------------------------------------------------
__global__ void __launch_bounds__(256) msg_kernel(const unsigned long long* __restrict__ adjbits,
                                                  const float* __restrict__ edges,
                                                  const float* __restrict__ P3,
                                                  const float* __restrict__ bmsg,
                                                  const float* __restrict__ Abuf,
                                                  const float* __restrict__ Cbuf,
                                                  float* __restrict__ msg) {
  const int bi = blockIdx.x;              // b*N + i
  const int b = bi / Nn;
  const int mt = threadIdx.x >> 5;
  const int lane = threadIdx.x & 31;
  const int lo = lane & 15, hi = lane >> 4;
  const int m = mt * 16 + lo;
  const float* B3 = P3 + (mt * 4) * 64 + lane * 2;   // K/4 = 4 frags
  const float a_i = Abuf[bi * MM + m];
  const float bm = bmsg[m];
  const unsigned long long abits = adjbits[bi];
  const float* erow = edges + (size_t)bi * Nn * EF;

  float partial = 0.0f;
  for (int jt = 0; jt < 3; ++jt) {
    v8f acc = {};
    const int ja = jt * 16 + lo;          // A-fragment row (j index)
    const bool jok = (ja < Nn);
    const float* ep = erow + ja * EF;
#pragma unroll
    for (int ks = 0; ks < 4; ++ks) {
      v2f a;
      if (jok) a = *(const v2f*)(ep + ks * 4 + 2 * hi);
      else { a.x = 0.0f; a.y = 0.0f; }
      v2f bb = *(const v2f*)(B3 + ks * 64);
      acc = wmma4(a, bb, acc);
    }
#pragma unroll
    for (int r = 0; r < 8; ++r) {
      const int j = jt * 16 + r + 8 * hi;
      if (j < Nn && ((abits >> j) & 1ull)) {
        const float v = acc[r] + a_i + Cbuf[(b * Nn + j) * MM + m] + bm;
        partial += fmaxf(v, 0.0f);
      }
    }
  }
  const float total = partial + __shfl_xor(partial, 16, 32);
  if (hi == 0) msg[bi * MM + m] = total;
}

// ---------------------------------------------------------------------------
// upd: Hout = mask ? relu([Hin | Msg] @ W_upd + b_upd) : Hin    (K = 256)
// ---------------------------------------------------------------------------
__global__ void __launch_bounds__(128) upd_kernel(const float* __restrict__ Hin,
                                                  const float* __restrict__ Msg,
                                                  const float* __restrict__ Pupd,
                                                  const float* __restrict__ bupd,
                                                  const float* __restrict__ maskf,
                                                  float* __restrict__ Hout) {
  const int lane = threadIdx.x & 31;
  const int wave = threadIdx.x >> 5;
  const int tile = blockIdx.x * 4 + wave;
  const int mt = tile >> 3;
  const int nt = tile & 7;
  const int lo = lane & 15, hi = lane >> 4;
  const int row = mt * 16 + lo;
  const int col = nt * 16 + lo;
  const float* B = Pupd + (nt * 64) * 64 + lane * 2;  // K/4 = 64 frags

  v8f acc = {};
#pragma unroll 8
  for (int ks = 0; ks < 64; ++ks) {
    const int k = ks * 4 + 2 * hi;
    v2f a = (k < NF) ? *(const v2f*)(Hin + row * NF + k)
                     : *(const v2f*)(Msg + row * MM + (k - NF));
    v2f bb = *(const v2f*)(B + ks * 64);
    acc = wmma4(a, bb, acc);
  }
  const float bu = bupd[col];
#pragma unroll
  for (int r = 0; r < 8; ++r) {
    const int orow = mt * 16 + r + 8 * hi;
    const float u = fmaxf(acc[r] + bu, 0.0f);
    Hout[orow * NF + col] = (maskf[orow] != 0.0f) ? u : Hin[orow * NF + col];
  }
}

// ---------------------------------------------------------------------------
// readout: out[b,o] = sum_n mask[b,n] * ([H|nodes][b,n,:] @ W_ro[:,o] + b_ro[o])
// One block per b; 8 waves = 8 o-tiles; n tiled 3x16 (padded, zero-guarded).
// ---------------------------------------------------------------------------
__global__ void __launch_bounds__(256) readout_kernel(const float* __restrict__ H,
                                                      const float* __restrict__ nodes,
                                                      const float* __restrict__ Pro,
                                                      const float* __restrict__ bro,
                                                      const float* __restrict__ maskf,
                                                      float* __restrict__ out) {
  const int b = blockIdx.x;
  const int ot = threadIdx.x >> 5;
  const int lane = threadIdx.x & 31;
  const int lo = lane & 15, hi = lane >> 4;
  const int o = ot * 16 + lo;
  const float bo = bro[o];
  const float* B = Pro + (ot * 64) * 64 + lane * 2;   // K/4 = 64 frags

  float partial = 0.0f;
  for (int nt = 0; nt < 3; ++nt) {
    v8f acc = {};
    const int na = nt * 16 + lo;
    const bool nok = (na < Nn);
    const int rowa = b * Nn + na;
#pragma unroll 8
    for (int ks = 0; ks < 64; ++ks) {
      const int k = ks * 4 + 2 * hi;
      v2f a;
      if (nok) {
        a = (k < NF) ? *(const v2f*)(H + rowa * NF + k)
                     : *(const v2f*)(nodes + rowa * NF + (k - NF));
      } else { a.x = 0.0f; a.y = 0.0f; }
      v2f bb = *(const v2f*)(B + ks * 64);
      acc = wmma4(a, bb, acc);
    }
#pragma unroll
    for (int r = 0; r < 8; ++r) {
      const int n = nt * 16 + r + 8 * hi;
      if (n < Nn) {
        partial += maskf[b * Nn + n] * (acc[r] + bo);
      }
    }
  }
  const float total = partial + __shfl_xor(partial, 16, 32);
  if (hi == 0) out[b * OUT + o] = total;
}

// ---------------------------------------------------------------------------
extern "C" void kernel_launch(void* const* d_in, const int* in_sizes, int n_in,
                              void* d_out, int out_size, void* d_ws, size_t ws_size,
                              hipStream_t stream) {
  const int*   adj   = (const int*)d_in[0];
  const float* nodes = (const float*)d_in[1];
  const float* edges = (const float*)d_in[2];
  const float* Wmsg  = (const float*)d_in[3];
  const float* bmsg  = (const float*)d_in[4];
  const float* Wupd  = (const float*)d_in[5];
  const float* bupd  = (const float*)d_in[6];
  const float* Wro   = (const float*)d_in[7];
  const float* bro   = (const float*)d_in[8];
  float* out = (float*)d_out;

  float* ws    = (float*)d_ws;
  float* h0    = ws;                  // [BN, NF]
  float* h1    = h0 + BN * NF;        // [BN, NF]
  float* Abuf  = h1 + BN * NF;        // [BN, MM]
  float* Cbuf  = Abuf + BN * MM;      // [BN, MM]
  float* msg   = Cbuf + BN * MM;      // [BN, MM]
  float* maskf = msg + BN * MM;       // [BN]
  unsigned long long* adjbits = (unsigned long long*)(maskf + BN);  // [BN]
  float* P1    = (float*)(adjbits + BN);  // [128*128] packed W1
  float* P2    = P1 + NF * MM;            // [128*128] packed W2
  float* P3    = P2 + NF * MM;            // [16*128]  packed W3
  float* Pupd  = P3 + EF * MM;            // [256*128]
  float* Pro   = Pupd + (NF + MM) * NF;   // [256*128]

  init_kernel<<<640, 256, 0, stream>>>(adj, nodes, h0, maskf, adjbits);
  pack_b_kernel<<<64, 256, 0, stream>>>(Wmsg,            P1,   NF,      MM);
  pack_b_kernel<<<64, 256, 0, stream>>>(Wmsg + NF * MM,  P2,   NF,      MM);
  pack_b_kernel<<<8,  256, 0, stream>>>(Wmsg + 2*NF*MM,  P3,   EF,      MM);
  pack_b_kernel<<<128,256, 0, stream>>>(Wupd,            Pupd, NF + MM, NF);
  pack_b_kernel<<<128,256, 0, stream>>>(Wro,             Pro,  2 * NF,  OUT);

  float* hin = h0;
  float* hout = h1;
  for (int p = 0; p < PASSES; ++p) {
    proj_kernel<<<640, 128, 0, stream>>>(hin, P1, P2, Abuf, Cbuf);
    msg_kernel<<<Bm * Nn, 256, 0, stream>>>(adjbits, edges, P3, bmsg, Abuf, Cbuf, msg);
    upd_kernel<<<640, 128, 0, stream>>>(hin, msg, Pupd, bupd, maskf, hout);
    float* t = hin; hin = hout; hout = t;
  }
  readout_kernel<<<Bm, 256, 0, stream>>>(hin, nodes, Pro, bro, maskf, out);
}